// Rule_12910671692595
// MI455X (gfx1250) — compile-verified
//
#include <hip/hip_runtime.h>
#include <cmath>
#include <cstdint>

#ifndef __has_builtin
#define __has_builtin(x) 0
#endif

#if __has_builtin(__builtin_amdgcn_global_load_async_to_lds_b32)
#define ATHENA_ASYNC_LDS 1
#else
#define ATHENA_ASYNC_LDS 0
#endif

typedef __attribute__((address_space(1))) int g_int;   // global
typedef __attribute__((address_space(3))) int l_int;   // LDS

#define TILE 32
#define R3W 38              // TILE+6 : (trail+mass) staging halo
#define R2W 36              // TILE+4 : trail_new region
#define R1W 34              // TILE+2 : prop / momentum_upd region
#define N3 (R3W*R3W)        // 1444
#define N2 (R2W*R2W)        // 1296
#define N1 (R1W*R1W)        // 1156
#define SQ2f 0.70710678118654752440f

__device__ __forceinline__ int wrapi(int v, int n) {
  if (v < 0) v += n;
  if (v >= n) v -= n;
  return v;
}

__global__ __launch_bounds__(256) void physarum_fused(
    const float* __restrict__ mass, const float* __restrict__ trail,
    const float* __restrict__ mom,
    float* __restrict__ o_mass, float* __restrict__ o_trail,
    float* __restrict__ o_mom, float* __restrict__ o_force,
    float* __restrict__ part_in, float* __restrict__ part_new,
    int H, int W)
{
  __shared__ float s_prop[9 * N1];   // first N3 floats aliased as s_tm in phase 0/1
  __shared__ float s_tn[N2];
  __shared__ float s_mass[N3];
  __shared__ float s_mu[2 * N1];
  __shared__ float s_red[256];
  float* s_tm = s_prop;              // trail+mass staging; dead before prop is written

  const int t  = threadIdx.x;
  const int by = blockIdx.y * TILE, bx = blockIdx.x * TILE;
  const size_t HW = (size_t)H * W;

  // ---- Phase 0: stage trail and mass over (TILE+6)^2 with circular wrap ----
  for (int idx = t; idx < N3; idx += 256) {
    int r = idx / R3W, c = idx - r * R3W;
    int gy = wrapi(by + r - 3, H);
    int gx = wrapi(bx + c - 3, W);
    size_t g = (size_t)gy * W + gx;
#if ATHENA_ASYNC_LDS
    __builtin_amdgcn_global_load_async_to_lds_b32(
        (g_int*)(trail + g), (l_int*)(s_tm + idx), 0, 0);
    __builtin_amdgcn_global_load_async_to_lds_b32(
        (g_int*)(mass + g), (l_int*)(s_mass + idx), 0, 0);
#else
    float mv = mass[g];
    s_mass[idx] = mv;
    s_tm[idx] = trail[g] + mv;
#endif
  }
#if ATHENA_ASYNC_LDS
#if __has_builtin(__builtin_amdgcn_s_wait_asynccnt)
  __builtin_amdgcn_s_wait_asynccnt(0);
#else
  asm volatile("s_wait_asynccnt 0" ::: "memory");
#endif
  __syncthreads();
  for (int idx = t; idx < N3; idx += 256) s_tm[idx] += s_mass[idx];
#endif
  __syncthreads();

  // ---- Phase 1: trail_new = (1-A) * 3x3 circular avg of (trail+mass) on (TILE+4)^2 ----
  for (int idx = t; idx < N2; idx += 256) {
    int r = idx / R2W, c = idx - r * R2W;
    const float* p0 = s_tm + r * R3W + c;
    float s = p0[0] + p0[1] + p0[2]
            + p0[R3W] + p0[R3W + 1] + p0[R3W + 2]
            + p0[2 * R3W] + p0[2 * R3W + 1] + p0[2 * R3W + 2];
    s_tn[idx] = s * (0.9f / 9.0f);
  }
  __syncthreads();

  // ---- Phase 2: force_vec, momentum update, velocity, 9-way softmax prop on (TILE+2)^2 ----
  const float DY[9] = { SQ2f, 0.f, -SQ2f, 1.f, 0.f, -1.f,  SQ2f,  0.f, -SQ2f };
  const float DX[9] = { SQ2f, 1.f,  SQ2f, 0.f, 0.f,  0.f, -SQ2f, -1.f, -SQ2f };
  for (int idx = t; idx < N1; idx += 256) {
    int r = idx / R1W, c = idx - r * R1W;
    const float* q0 = s_tn + r * R2W + c;            // tn(top-left of 3x3 around this pos)
    float f0 = q0[0], f1 = q0[1], f2 = q0[2];
    float f3 = q0[R2W], f5 = q0[R2W + 2];
    float f6 = q0[2 * R2W], f7 = q0[2 * R2W + 1], f8 = q0[2 * R2W + 2];
    float fy = SQ2f * (f0 - f2 + f6 - f8) + (f3 - f5);
    float fx = SQ2f * (f0 + f2 - f6 - f8) + (f1 - f7);

    float m = s_mass[(r + 2) * R3W + (c + 2)];
    bool  mk = m < 1e-8f;
    float sm = mk ? 1.0f : m;
    int gy = wrapi(by + r - 1, H);
    int gx = wrapi(bx + c - 1, W);
    size_t g = (size_t)gy * W + gx;
    float muy = mk ? 0.0f : fmaf(fy, 0.1f, mom[g]);
    float mux = mk ? 0.0f : fmaf(fx, 0.1f, mom[HW + g]);
    float vy = mk ? 0.0f : muy / sm;
    float vx = mk ? 0.0f : mux / sm;

    float z[9];
    float mx = -3.402823e38f;
#pragma unroll
    for (int d = 0; d < 9; ++d) {
      float dv = DY[d] * vy + DX[d] * vx;
      dv = (dv <= 0.0f) ? 1e-10f : dv;
      dv *= 10.0f;                      // temp
      z[d] = dv;
      mx = fmaxf(mx, dv);
    }
    float se = 0.0f;
#pragma unroll
    for (int d = 0; d < 9; ++d) { float e = expf(z[d] - mx); z[d] = e; se += e; }
    float inv = 1.0f / se;
#pragma unroll
    for (int d = 0; d < 9; ++d) s_prop[d * N1 + idx] = z[d] * inv;
    s_mu[idx]      = muy;
    s_mu[N1 + idx] = mux;
  }
  __syncthreads();

  // ---- Phase 3: outputs. Each thread handles 4 consecutive x (float4 stores). ----
  const int ty = t >> 3;            // 0..31
  const int x0 = (t & 7) * 4;       // 0..28
  const size_t obase = (size_t)(by + ty) * W + (bx + x0);

  { // trail out
    const float* p = s_tn + (ty + 2) * R2W + (x0 + 2);
    *(float4*)(o_trail + obase) = make_float4(p[0], p[1], p[2], p[3]);
  }
  // force out: force[d] = trail_new at unfold offset (d/3-1, d%3-1)
#pragma unroll
  for (int d = 0; d < 9; ++d) {
    const float* p = s_tn + (ty + 2 + (d / 3) - 1) * R2W + (x0 + 2 + (d % 3) - 1);
    *(float4*)(o_force + (size_t)d * HW + obase) = make_float4(p[0], p[1], p[2], p[3]);
  }
  // "moves" gathers: offset is the negative of the unfold offset
  float accm[4] = {0.f, 0.f, 0.f, 0.f};
  float accy[4] = {0.f, 0.f, 0.f, 0.f};
  float accx[4] = {0.f, 0.f, 0.f, 0.f};
#pragma unroll
  for (int d = 0; d < 9; ++d) {
    int qr  = ty + 2 - d / 3;       // R1 row of source pixel
    int qc0 = x0 + 2 - d % 3;       // R1 col base
    const float* pp = s_prop + d * N1 + qr * R1W + qc0;
    const float* pm = s_mass + (qr + 2) * R3W + (qc0 + 2);
    const float* py = s_mu + qr * R1W + qc0;
    const float* px = s_mu + N1 + qr * R1W + qc0;
#pragma unroll
    for (int j = 0; j < 4; ++j) {
      float p = pp[j];
      accm[j] = fmaf(pm[j], p, accm[j]);
      accy[j] = fmaf(py[j], p, accy[j]);
      accx[j] = fmaf(px[j], p, accx[j]);
    }
  }
  *(float4*)(o_mass + obase)      = make_float4(accm[0], accm[1], accm[2], accm[3]);
  *(float4*)(o_mom + obase)       = make_float4(accy[0], accy[1], accy[2], accy[3]);
  *(float4*)(o_mom + HW + obase)  = make_float4(accx[0], accx[1], accx[2], accx[3]);

  // ---- deterministic block partial sums (no float atomics) ----
  float lnew = accm[0] + accm[1] + accm[2] + accm[3];
  const float* pmc = s_mass + (ty + 3) * R3W + (x0 + 3);
  float lin = pmc[0] + pmc[1] + pmc[2] + pmc[3];

  int bid = blockIdx.y * gridDim.x + blockIdx.x;
  s_red[t] = lin; __syncthreads();
  for (int s = 128; s > 0; s >>= 1) { if (t < s) s_red[t] += s_red[t + s]; __syncthreads(); }
  if (t == 0) part_in[bid] = s_red[0];
  __syncthreads();
  s_red[t] = lnew; __syncthreads();
  for (int s = 128; s > 0; s >>= 1) { if (t < s) s_red[t] += s_red[t + s]; __syncthreads(); }
  if (t == 0) part_new[bid] = s_red[0];
}

__global__ __launch_bounds__(256) void reduce_partials(
    const float* __restrict__ part_in, const float* __restrict__ part_new,
    float* __restrict__ sums, int n)
{
  __shared__ float sh[256];
  int t = threadIdx.x;
  float a = 0.f, b = 0.f;
  for (int i = t; i < n; i += 256) { a += part_in[i]; b += part_new[i]; }
  sh[t] = a; __syncthreads();
  for (int s = 128; s > 0; s >>= 1) { if (t < s) sh[t] += sh[t + s]; __syncthreads(); }
  if (t == 0) sums[0] = sh[0];
  __syncthreads();
  sh[t] = b; __syncthreads();
  for (int s = 128; s > 0; s >>= 1) { if (t < s) sh[t] += sh[t + s]; __syncthreads(); }
  if (t == 0) sums[1] = sh[0];
}

__global__ __launch_bounds__(256) void scale_mass(
    float* __restrict__ o_mass, const float* __restrict__ sums, int n4)
{
  int i = blockIdx.x * blockDim.x + threadIdx.x;
  if (i >= n4) return;
  float sc = sums[0] / (sums[1] + 1e-10f);
  float4 v = ((float4*)o_mass)[i];
  v.x *= sc; v.y *= sc; v.z *= sc; v.w *= sc;
  ((float4*)o_mass)[i] = v;
}

extern "C" void kernel_launch(void* const* d_in, const int* in_sizes, int n_in,
                              void* d_out, int out_size, void* d_ws, size_t ws_size,
                              hipStream_t stream) {
  (void)n_in; (void)out_size; (void)ws_size;
  const float* mass  = (const float*)d_in[0];
  const float* trail = (const float*)d_in[1];
  const float* mom   = (const float*)d_in[2];

  int HW = in_sizes[0];
  int H = (int)lround(sqrt((double)HW));
  int W = H;

  float* out    = (float*)d_out;
  float* o_mass  = out;
  float* o_trail = out + (size_t)HW;
  float* o_mom   = out + (size_t)2 * HW;
  float* o_force = out + (size_t)4 * HW;

  float* ws       = (float*)d_ws;
  float* sums     = ws;          // [0]=sum(mass_in), [1]=sum(mass_new_un)
  int nbx = W / TILE, nby = H / TILE;
  int nblk = nbx * nby;
  float* part_in  = ws + 16;
  float* part_new = part_in + nblk;

  dim3 grid(nbx, nby);
  physarum_fused<<<grid, 256, 0, stream>>>(mass, trail, mom,
                                           o_mass, o_trail, o_mom, o_force,
                                           part_in, part_new, H, W);
  reduce_partials<<<1, 256, 0, stream>>>(part_in, part_new, sums, nblk);
  int n4 = HW / 4;
  scale_mass<<<(n4 + 255) / 256, 256, 0, stream>>>(o_mass, sums, n4);
}